// ContinuousWaveletLayer_33028298506965
// MI455X (gfx1250) — compile-verified
//
#include <hip/hip_runtime.h>
#include <hip/hip_bf16.h>

typedef float v2f __attribute__((ext_vector_type(2)));
typedef float v8f __attribute__((ext_vector_type(8)));
typedef int   v4i __attribute__((ext_vector_type(4)));

#define NSC    32          // number of scales
#define LMAX   256         // padded wavelet length
#define NTIME  8192
#define NBATCH 128
#define CHUNK  2048        // time points per block
#define NCHUNK (NTIME / CHUNK)   // 4
#define BLK    256               // 8 waves (wave32)
#define NWAVE  (BLK / 32)
#define XWIN   (CHUNK + LMAX)    // 2304 window incl. halo

// workspace layout (in floats)
#define WS_W    0                       // [0, 8192)   reversed wavelet bank W[s][l]
#define WS_SOFT (NSC * LMAX)            // [8192, 8224) softmax(scale_weights)
#define WS_PART (WS_SOFT + NSC)         // [8224, 8224 + 128*4*32) per-block partials

// gfx1250 async global->LDS path (ASYNCcnt-tracked, bypasses VGPRs).
// Param types from compiler diagnostics: (AS1 typed ptr, AS3 typed ptr, imm, imm).
#if __has_builtin(__builtin_amdgcn_global_load_async_to_lds_b128) && \
    __has_builtin(__builtin_amdgcn_global_load_async_to_lds_b32)
#define USE_ASYNC_LDS 1
typedef __attribute__((address_space(1))) v4i* g128_t;
typedef __attribute__((address_space(3))) v4i* l128_t;
typedef __attribute__((address_space(1))) int* g32_t;
typedef __attribute__((address_space(3))) int* l32_t;
#else
#define USE_ASYNC_LDS 0
#endif

// ---------------------------------------------------------------------------
// Kernel 1: build W[s][l] = bank[s][255-l] (flipped Morlet bank) + softmax.
// ---------------------------------------------------------------------------
__global__ void cwt_prep(const float* __restrict__ sw, float* __restrict__ ws) {
    const int i = blockIdx.x;        // scale index 0..31
    const int l = threadIdx.x;       // tap index 0..255
    const int s = i + 1;
    const float sf = (float)s;
    const int L   = 8 * s;           // min(8*s, 8192) == 8*s here
    const int pos = (LMAX - 1) - l;  // bank position (flip for true convolution)
    const int off = 128 - 4 * s;     // (Lmax-1)/2 - (L-1)/2
    const int j   = pos - off;
    float val = 0.0f;
    if (j >= 0 && j < L) {
        float t = -4.0f * sf + (8.0f * sf) * (float)j / (float)(L - 1);
        val = expf(-t * t / (2.0f * sf * sf)) * cosf(5.0f * t / sf) * rsqrtf(sf);
    }
    ws[WS_W + i * LMAX + l] = val;

    if (i == 0 && l == 0) {          // serial softmax over 32 weights (tiny)
        float m = sw[0];
        for (int k = 1; k < NSC; ++k) m = fmaxf(m, sw[k]);
        float e[NSC]; float sum = 0.0f;
        for (int k = 0; k < NSC; ++k) { e[k] = expf(sw[k] - m); sum += e[k]; }
        const float inv = 1.0f / sum;
        for (int k = 0; k < NSC; ++k) ws[WS_SOFT + k] = e[k] * inv;
    }
}

// ---------------------------------------------------------------------------
// Kernel 2: implicit-GEMM CWT energy via V_WMMA_F32_16X16X4_F32.
//   coef[b,s,t] = sum_l x[b, t-128+l] * W[s,l]   (M=time tile 16, N=16, K=4)
//   partial[b,chunk,s] = sum_{t in chunk} coef^2
// ---------------------------------------------------------------------------
__global__ void __launch_bounds__(BLK) cwt_main(const float* __restrict__ x,
                                                const float* __restrict__ ws,
                                                float* __restrict__ part) {
    __shared__ __align__(16) float sW[NSC * LMAX];   // 32 KB wavelet bank
    __shared__ __align__(16) float sX[XWIN];         // 9 KB x window (+-128 halo)
    __shared__ float sRed[2 * BLK];                  // reduction scratch

    const int chunk = blockIdx.x;
    const int b     = blockIdx.y;
    const int tid   = threadIdx.x;
    const int c0    = chunk * CHUNK;

#if USE_ASYNC_LDS
    {
        // 32 KB bank: 128-bit async copies, both sides 16B aligned
        const float* gW = ws + WS_W;
        for (int i = tid * 4; i < NSC * LMAX; i += BLK * 4)
            __builtin_amdgcn_global_load_async_to_lds_b128(
                (g128_t)(gW + i), (l128_t)&sW[i], 0, 0);
        // x window: per-lane 32-bit async loads, EXEC-predicated halo
        const int basei = c0 - (LMAX / 2);
        const float* gx = x + (size_t)b * NTIME;
        for (int j = tid; j < XWIN; j += BLK) {
            const int idx = basei + j;
            if (idx >= 0 && idx < NTIME)
                __builtin_amdgcn_global_load_async_to_lds_b32(
                    (g32_t)(gx + idx), (l32_t)&sX[j], 0, 0);
            else
                sX[j] = 0.0f;
        }
#if __has_builtin(__builtin_amdgcn_s_wait_asynccnt)
        __builtin_amdgcn_s_wait_asynccnt(0);
#else
        asm volatile("s_wait_asynccnt 0" ::: "memory");
#endif
    }
#else
    for (int i = tid; i < NSC * LMAX; i += BLK) sW[i] = ws[WS_W + i];
    for (int j = tid; j < XWIN; j += BLK) {
        const int idx = c0 - (LMAX / 2) + j;                 // x index t-128
        sX[j] = (idx >= 0 && idx < NTIME) ? x[(size_t)b * NTIME + idx] : 0.0f;
    }
#endif
    __syncthreads();

    const int wave = tid >> 5;
    const int lane = tid & 31;
    const int col  = lane & 15;   // A: row M; B/D: column N
    const int half = lane >> 4;   // selects K pair {2h, 2h+1}

    float e0 = 0.0f, e1 = 0.0f;   // per-lane energy for scale col and col+16

    const float* wp0 = &sW[col * LMAX];          // W row for scales 0..15
    const float* wp1 = &sW[(16 + col) * LMAX];   // W row for scales 16..31

    const int tilesTotal = CHUNK / 16;           // 128 M-tiles; uniform per wave
    for (int tile = wave; tile < tilesTotal; tile += NWAVE) {
        const int base = tile * 16 + col;        // sX offset of A row start
        v8f acc0 = {0.f, 0.f, 0.f, 0.f, 0.f, 0.f, 0.f, 0.f};
        v8f acc1 = {0.f, 0.f, 0.f, 0.f, 0.f, 0.f, 0.f, 0.f};

        #pragma unroll 4
        for (int kk = 0; kk < LMAX / 4; ++kk) {  // 64 k-steps of K=4
            const int ka = 4 * kk + 2 * half;
            v2f a;                                // A[m][k]: sliding window of x
            a.x = sX[base + ka];
            a.y = sX[base + ka + 1];
            const float2 w0 = *(const float2*)(wp0 + ka);   // 8B-aligned
            const float2 w1 = *(const float2*)(wp1 + ka);
            v2f b0; b0.x = w0.x; b0.y = w0.y;
            v2f b1; b1.x = w1.x; b1.y = w1.y;
            // D = A*B + C, full f32: v_wmma_f32_16x16x4_f32
            acc0 = __builtin_amdgcn_wmma_f32_16x16x4_f32(
                false, a, false, b0, (short)0, acc0, false, false);
            acc1 = __builtin_amdgcn_wmma_f32_16x16x4_f32(
                false, a, false, b1, (short)0, acc1, false, false);
        }
        // epilogue: all 8 D elements in this lane belong to one scale column
        #pragma unroll
        for (int i = 0; i < 8; ++i) {
            e0 += acc0[i] * acc0[i];
            e1 += acc1[i] * acc1[i];
        }
    }

    sRed[tid]       = e0;
    sRed[BLK + tid] = e1;
    __syncthreads();

    if (tid < NSC) {                 // per-block partial for each scale
        const int n = tid >> 4;      // which N-tile
        const int c = tid & 15;      // column within tile
        float sum = 0.0f;
        #pragma unroll
        for (int w = 0; w < NWAVE; ++w)
            sum += sRed[n * BLK + w * 32 + c] + sRed[n * BLK + w * 32 + c + 16];
        part[(b * NCHUNK + chunk) * NSC + tid] = sum;
    }
}

// ---------------------------------------------------------------------------
// Kernel 3: deterministic cross-chunk combine + softmax scale + mean.
// ---------------------------------------------------------------------------
__global__ void cwt_final(const float* __restrict__ ws, float* __restrict__ out) {
    const int t = blockIdx.x * blockDim.x + threadIdx.x;
    if (t >= NBATCH * NSC) return;
    const int b = t / NSC;
    const int s = t % NSC;
    float sum = 0.0f;
    #pragma unroll
    for (int c = 0; c < NCHUNK; ++c)
        sum += ws[WS_PART + (b * NCHUNK + c) * NSC + s];
    out[t] = sum * ws[WS_SOFT + s] * (1.0f / (float)NTIME);
}

extern "C" void kernel_launch(void* const* d_in, const int* in_sizes, int n_in,
                              void* d_out, int out_size, void* d_ws, size_t ws_size,
                              hipStream_t stream) {
    (void)in_sizes; (void)n_in; (void)out_size; (void)ws_size;
    const float* x  = (const float*)d_in[0];   // [128, 8192] f32
    const float* sw = (const float*)d_in[1];   // [32] f32
    float* out = (float*)d_out;                // [128, 32] f32
    float* ws  = (float*)d_ws;

    cwt_prep<<<NSC, LMAX, 0, stream>>>(sw, ws);
    dim3 grid(NCHUNK, NBATCH);                 // 512 blocks x 256 threads
    cwt_main<<<grid, BLK, 0, stream>>>(x, ws, ws + WS_PART);
    cwt_final<<<(NBATCH * NSC + 255) / 256, 256, 0, stream>>>(ws, out);
}